// BBoxDecoder_83056077570233
// MI455X (gfx1250) — compile-verified
//
#include <hip/hip_runtime.h>
#include <hip/hip_bf16.h>
#include <math.h>

typedef __attribute__((ext_vector_type(2))) float v2f;
typedef __attribute__((ext_vector_type(8))) float v8f;

// ---------------------------------------------------------------------------
// Generic strided f32 WMMA GEMM: C[M,N] = op(A[M,K] * B[K,N] + bias) * colscale
// One wave (32 threads) per block; each wave computes a 16(M) x 64(N) tile
// with V_WMMA_F32_16X16X4_F32, K-loop in steps of 4 (A frag reused across the
// 4 N-subtiles -> 4 back-to-back wmmas per step).
//
// A element (m,k) at A[m*as_m + k*as_k]; B element (k,n) at B[k*bs_k + n*bs_n].
// All offsets are 32-bit ints against uniform base pointers so the backend can
// use the SADDR + VGPR-offset global-load form instead of 64-bit VGPR address
// pairs. Template flags:
//   BIAS/RELU/SCALE : straight-line fused epilogue
//   AK1  : as_k == 1 -> A fragment is one b64 load
//   BN1  : bs_n == 1 -> N-subtile offsets are compile-time (fold to imm offs)
//   BK1  : bs_k == 1 -> B fragments are b64 loads
// ---------------------------------------------------------------------------
template <bool BIAS, bool RELU, bool SCALE, bool AK1, bool BN1, bool BK1>
__global__ __launch_bounds__(32) void gemm_wmma_f32(
    const float* __restrict__ A, int as_m, int as_k,
    const float* __restrict__ B, int bs_k, int bs_n,
    float* __restrict__ C, int M, int N, int K,
    const float* __restrict__ bias, const float* __restrict__ colscale)
{
    const int tiles_n = N >> 6;                 // N/64
    const int tm = (int)blockIdx.x / tiles_n;
    const int tn = (int)blockIdx.x % tiles_n;
    const int m0 = tm << 4;                     // 16-row tile
    const int n0 = tn << 6;                     // 64-col tile

    const int lane = threadIdx.x;               // wave32
    const int half = lane >> 4;                 // 0: K pair {0,1}, 1: {2,3}
    const int l16  = lane & 15;

    const int m  = m0 + l16;                    // A-matrix row held by this lane
    const int nb = n0 + l16;                    // base column for B frags

    // Per-lane 32-bit element offsets, bumped by 4*stride each K-step.
    int aoff = m * as_m + (half << 1) * as_k;
    const int bnstride = BN1 ? 1 : bs_n;
    int boff = nb * bnstride + (half << 1) * bs_k;
    const int bcol = BN1 ? 16 : 16 * bs_n;      // N-subtile step (const if BN1)
    const int a_step = AK1 ? 4 : 4 * as_k;
    const int b_step = BK1 ? 4 : 4 * bs_k;

    v8f acc0 = {}, acc1 = {}, acc2 = {}, acc3 = {};

    for (int k0 = 0; k0 < K; k0 += 4) {
        v2f a;
        if (AK1) {
            a = *(const v2f*)(A + aoff);        // one global_load_b64
        } else {
            a.x = A[aoff];
            a.y = A[aoff + as_k];
        }

        v2f b0, b1, b2, b3;
        if (BK1) {
            b0 = *(const v2f*)(B + boff);
            b1 = *(const v2f*)(B + boff + bcol);
            b2 = *(const v2f*)(B + boff + 2 * bcol);
            b3 = *(const v2f*)(B + boff + 3 * bcol);
        } else {
            b0.x = B[boff];                b0.y = B[boff + bs_k];
            b1.x = B[boff + bcol];         b1.y = B[boff + bcol + bs_k];
            b2.x = B[boff + 2 * bcol];     b2.y = B[boff + 2 * bcol + bs_k];
            b3.x = B[boff + 3 * bcol];     b3.y = B[boff + 3 * bcol + bs_k];
        }

        acc0 = __builtin_amdgcn_wmma_f32_16x16x4_f32(false, a, false, b0, (short)0, acc0, false, false);
        acc1 = __builtin_amdgcn_wmma_f32_16x16x4_f32(false, a, false, b1, (short)0, acc1, false, false);
        acc2 = __builtin_amdgcn_wmma_f32_16x16x4_f32(false, a, false, b2, (short)0, acc2, false, false);
        acc3 = __builtin_amdgcn_wmma_f32_16x16x4_f32(false, a, false, b3, (short)0, acc3, false, false);

        aoff += a_step;
        boff += b_step;
    }

    // Epilogue. C/D layout: VGPR v -> m = v + 8*half, n = (lane&15) in subtile.
    #pragma unroll
    for (int t = 0; t < 4; ++t) {
        const int nn = n0 + (t << 4) + l16;
        const v8f acc = (t == 0) ? acc0 : (t == 1) ? acc1 : (t == 2) ? acc2 : acc3;
        const float bval = BIAS  ? bias[nn]     : 0.0f;
        const float sval = SCALE ? colscale[nn] : 1.0f;
        int coff = (m0 + (half << 3)) * N + nn;
        #pragma unroll
        for (int v = 0; v < 8; ++v) {
            float val = acc[v];
            if (BIAS)  val += bval;
            if (RELU)  val  = fmaxf(val, 0.0f);
            if (SCALE) val *= sval;
            C[coff] = val;
            coff += N;
        }
    }
}

// ---------------------------------------------------------------------------
// mean over pixels per channel: mean[c] = (1/P) * sum_p enc[c*P + p]
// ---------------------------------------------------------------------------
__global__ __launch_bounds__(256) void mean_kernel(
    const float* __restrict__ enc, float* __restrict__ mean, int P)
{
    __shared__ float red[256];
    const int c = blockIdx.x;
    float s = 0.0f;
    for (int p = threadIdx.x; p < P; p += 256) s += enc[c * P + p];
    red[threadIdx.x] = s;
    __syncthreads();
    for (int stride = 128; stride > 0; stride >>= 1) {
        if (threadIdx.x < stride) red[threadIdx.x] += red[threadIdx.x + stride];
        __syncthreads();
    }
    if (threadIdx.x == 0) mean[c] = red[0] / (float)P;
}

// ---------------------------------------------------------------------------
// Small serial-chain vectors in one block (C = D = 512 threads):
//   h0[d]     = b_init[d] + mean . W_init[:,d]
//   bL[a]     = b_tag[a] + b_lang[a] + h0 . W_lang[:,a]     (a < 256)
//   gscale[c] = sigmoid(b_fbeta[c] + h0 . W_fbeta[:,c]) * h0[c]
// ---------------------------------------------------------------------------
__global__ __launch_bounds__(512) void vec_kernel(
    const float* __restrict__ mean,
    const float* __restrict__ W_init, const float* __restrict__ b_init,
    const float* __restrict__ W_lang, const float* __restrict__ b_lang,
    const float* __restrict__ b_tag,
    const float* __restrict__ W_fbeta, const float* __restrict__ b_fbeta,
    float* __restrict__ bL, float* __restrict__ gscale,
    int C, int D, int A)
{
    __shared__ float sm[512];
    __shared__ float sh[512];
    const int tid = threadIdx.x;

    sm[tid] = mean[tid];
    __syncthreads();

    float h = b_init[tid];
    for (int c = 0; c < C; ++c) h += sm[c] * W_init[c * D + tid];
    sh[tid] = h;
    __syncthreads();

    if (tid < A) {
        float l = b_lang[tid] + b_tag[tid];
        for (int d = 0; d < D; ++d) l += sh[d] * W_lang[d * A + tid];
        bL[tid] = l;
    }
    float g = b_fbeta[tid];
    for (int d = 0; d < D; ++d) g += sh[d] * W_fbeta[d * C + tid];
    g = 1.0f / (1.0f + __expf(-g));
    gscale[tid] = g * sh[tid];
}

// ---------------------------------------------------------------------------
// Additive-attention scores (relu inside the contraction -> VALU, not WMMA):
//   Z[n,p] = b_full + sum_a relu(RL[n,a] + S[p,a]) * w[a]
// Block computes a 64x64 (n,p) tile; thread holds a 4x4 register block;
// A-dim streamed through LDS in chunks of 32 (rows padded to 33 floats so the
// 16 tx-lanes hit 16 distinct banks instead of one).
// ---------------------------------------------------------------------------
#define SC_TILE 64
#define SC_KA   32
#define SC_PAD  33
__global__ __launch_bounds__(256) void score_kernel(
    const float* __restrict__ RL, const float* __restrict__ S,
    const float* __restrict__ w_full, const float* __restrict__ b_full,
    float* __restrict__ Z, int N, int P, int A)
{
    __shared__ float lr[SC_TILE][SC_PAD];
    __shared__ float ls[SC_TILE][SC_PAD];
    __shared__ float lw[SC_KA];

    const int n0 = blockIdx.y * SC_TILE;
    const int p0 = blockIdx.x * SC_TILE;
    const int tid = threadIdx.x;
    const int tx = tid & 15;        // p group
    const int ty = tid >> 4;        // n group

    float acc[4][4] = {};

    for (int a0 = 0; a0 < A; a0 += SC_KA) {
        // 64 rows x 32 cols = 512 float4 per matrix; 256 threads -> 2 each.
        for (int i = tid; i < SC_TILE * (SC_KA / 4); i += 256) {
            const int r = i >> 3;               // / (SC_KA/4)
            const int c4 = (i & 7) << 2;        // float4 column
            const float4 vr = *(const float4*)&RL[(n0 + r) * A + a0 + c4];
            const float4 vs = *(const float4*)&S [(p0 + r) * A + a0 + c4];
            lr[r][c4 + 0] = vr.x; lr[r][c4 + 1] = vr.y;
            lr[r][c4 + 2] = vr.z; lr[r][c4 + 3] = vr.w;
            ls[r][c4 + 0] = vs.x; ls[r][c4 + 1] = vs.y;
            ls[r][c4 + 2] = vs.z; ls[r][c4 + 3] = vs.w;
        }
        if (tid < SC_KA) lw[tid] = w_full[a0 + tid];
        __syncthreads();

        for (int a = 0; a < SC_KA; ++a) {
            const float wa = lw[a];
            float rr[4], ss[4];
            #pragma unroll
            for (int i = 0; i < 4; ++i) rr[i] = lr[ty * 4 + i][a];
            #pragma unroll
            for (int j = 0; j < 4; ++j) ss[j] = ls[tx * 4 + j][a];
            #pragma unroll
            for (int i = 0; i < 4; ++i)
                #pragma unroll
                for (int j = 0; j < 4; ++j)
                    acc[i][j] += fmaxf(rr[i] + ss[j], 0.0f) * wa;
        }
        __syncthreads();
    }

    const float bf = b_full[0];
    #pragma unroll
    for (int i = 0; i < 4; ++i) {
        const int row = (n0 + ty * 4 + i) * P + p0;
        #pragma unroll
        for (int j = 0; j < 4; ++j)
            Z[row + tx * 4 + j] = acc[i][j] + bf;
    }
}

// ---------------------------------------------------------------------------
// In-place row softmax over P elements; one block per row.
// ---------------------------------------------------------------------------
__global__ __launch_bounds__(256) void softmax_rows(float* __restrict__ Z, int P)
{
    __shared__ float red[256];
    float* row = Z + (long)blockIdx.x * P;
    const int tid = threadIdx.x;

    float mx = -INFINITY;
    for (int p = tid; p < P; p += 256) mx = fmaxf(mx, row[p]);
    red[tid] = mx;
    __syncthreads();
    for (int s = 128; s > 0; s >>= 1) {
        if (tid < s) red[tid] = fmaxf(red[tid], red[tid + s]);
        __syncthreads();
    }
    mx = red[0];
    __syncthreads();

    float sum = 0.0f;
    for (int p = tid; p < P; p += 256) {
        const float e = __expf(row[p] - mx);
        row[p] = e;
        sum += e;
    }
    red[tid] = sum;
    __syncthreads();
    for (int s = 128; s > 0; s >>= 1) {
        if (tid < s) red[tid] += red[tid + s];
        __syncthreads();
    }
    const float inv = 1.0f / red[0];
    for (int p = tid; p < P; p += 256) row[p] *= inv;
}

// ---------------------------------------------------------------------------
// Narrow heads (O = 31 or 4): one thread per (n, o), optional sigmoid.
// ---------------------------------------------------------------------------
__global__ void head_kernel(
    const float* __restrict__ X, const float* __restrict__ W,
    const float* __restrict__ b, float* __restrict__ out,
    int N, int C, int O, int do_sigmoid)
{
    const int idx = blockIdx.x * blockDim.x + threadIdx.x;
    if (idx >= N * O) return;
    const int n = idx / O, o = idx % O;
    float acc = b[o];
    const float* xr = X + n * C;
    for (int c = 0; c < C; ++c) acc += xr[c] * W[c * O + o];
    if (do_sigmoid) acc = 1.0f / (1.0f + __expf(-acc));
    out[idx] = acc;
}

// ---------------------------------------------------------------------------
extern "C" void kernel_launch(void* const* d_in, const int* in_sizes, int n_in,
                              void* d_out, int out_size, void* d_ws, size_t ws_size,
                              hipStream_t stream)
{
    (void)in_sizes; (void)n_in; (void)out_size; (void)ws_size;

    const int C = 512, P = 1024, A = 256, D = 512, N = 1024;
    const int O_CLS = 31, O_BOX = 4;

    const float* enc     = (const float*)d_in[0];   // [C, P] (NCHW flattened)
    const float* tag_H   = (const float*)d_in[1];   // [N, D]
    const float* W_enc   = (const float*)d_in[2];
    const float* b_enc   = (const float*)d_in[3];
    const float* W_tag   = (const float*)d_in[4];
    const float* b_tag   = (const float*)d_in[5];
    const float* W_lang  = (const float*)d_in[6];
    const float* b_lang  = (const float*)d_in[7];
    const float* w_full  = (const float*)d_in[8];
    const float* b_full  = (const float*)d_in[9];
    const float* W_init  = (const float*)d_in[10];
    const float* b_init  = (const float*)d_in[11];
    const float* W_fbeta = (const float*)d_in[12];
    const float* b_fbeta = (const float*)d_in[13];
    const float* W_cls   = (const float*)d_in[14];
    const float* b_cls   = (const float*)d_in[15];
    const float* W_m1    = (const float*)d_in[16];
    const float* b_m1    = (const float*)d_in[17];
    const float* W_m2    = (const float*)d_in[18];
    const float* b_m2    = (const float*)d_in[19];
    const float* W_m3    = (const float*)d_in[20];
    const float* b_m3    = (const float*)d_in[21];

    float* out = (float*)d_out;                    // [N*31] logits ++ [N*4] boxes

    // Workspace layout (floats)
    float* ws   = (float*)d_ws;
    float* mean = ws;               ws += 512;           // [C]
    float* bL   = ws;               ws += 256;           // [A] combined b_tag + L
    float* gsc  = ws;               ws += 512;           // [C] gate*h0
    float* Sp   = ws;               ws += (long)P * A;   // [P, A]
    float* RL   = ws;               ws += (long)N * A;   // [N, A]
    float* Z    = ws;               ws += (long)N * P;   // [N, P] -> alpha in place
    float* hbuf = ws;               ws += (long)N * C;   // [N, C]
    float* x1   = ws;               ws += (long)N * 256; // [N, 256]
    float* x2   = ws;               /* ws += N*256; */   // [N, 256]

    // 1) mean over pixels per channel
    mean_kernel<<<C, 256, 0, stream>>>(enc, mean, P);

    // 2) h0 chain -> combined bias bL, column scale gscale
    vec_kernel<<<1, 512, 0, stream>>>(mean, W_init, b_init, W_lang, b_lang,
                                      b_tag, W_fbeta, b_fbeta, bL, gsc, C, D, A);

    // 3) S_p = enc_flat @ W_enc + b_enc   (enc_flat[p,c] = enc[c*P + p])
    gemm_wmma_f32<true, false, false, false, true, false>
        <<<(P / 16) * (A / 64), 32, 0, stream>>>(
        enc, 1, P, W_enc, A, 1, Sp, P, A, C, b_enc, nullptr);

    // 4) RL = tag_H @ W_tag + (b_tag + L)
    gemm_wmma_f32<true, false, false, true, true, false>
        <<<(N / 16) * (A / 64), 32, 0, stream>>>(
        tag_H, D, 1, W_tag, A, 1, RL, N, A, D, bL, nullptr);

    // 5) Z[n,p] = b_full + sum_a relu(RL[n,a] + S_p[p,a]) * w_full[a]
    {
        dim3 grid(P / SC_TILE, N / SC_TILE);
        score_kernel<<<grid, 256, 0, stream>>>(RL, Sp, w_full, b_full, Z, N, P, A);
    }

    // 6) alpha = softmax(Z, axis=1), in place
    softmax_rows<<<N, 256, 0, stream>>>(Z, P);

    // 7) h = (alpha @ enc_flat) * (gate * h0)   (enc_flat[p,c] = enc[c*P + p])
    gemm_wmma_f32<false, false, true, true, false, true>
        <<<(N / 16) * (C / 64), 32, 0, stream>>>(
        Z, P, 1, enc, 1, P, hbuf, N, C, P, nullptr, gsc);

    // 8) x1 = relu(h @ W_m1 + b_m1)
    gemm_wmma_f32<true, true, false, true, true, false>
        <<<(N / 16) * (256 / 64), 32, 0, stream>>>(
        hbuf, C, 1, W_m1, 256, 1, x1, N, 256, C, b_m1, nullptr);

    // 9) x2 = relu(x1 @ W_m2 + b_m2)
    gemm_wmma_f32<true, true, false, true, true, false>
        <<<(N / 16) * (256 / 64), 32, 0, stream>>>(
        x1, 256, 1, W_m2, 256, 1, x2, N, 256, 256, b_m2, nullptr);

    // 10) logits = h @ W_cls + b_cls
    head_kernel<<<(N * O_CLS + 255) / 256, 256, 0, stream>>>(
        hbuf, W_cls, b_cls, out, N, C, O_CLS, 0);

    // 11) boxes = sigmoid(x2 @ W_m3 + b_m3)
    head_kernel<<<(N * O_BOX + 255) / 256, 256, 0, stream>>>(
        x2, W_m3, b_m3, out + N * O_CLS, N, 256, O_BOX, 1);
}